// BA3Net_72069551226970
// MI455X (gfx1250) — compile-verified
//
#include <hip/hip_runtime.h>
#include <hip/hip_bf16.h>

typedef __attribute__((ext_vector_type(2))) float v2f;
typedef __attribute__((ext_vector_type(8))) float v8f;

#define D 64
#define NGRAPH 256

__device__ __forceinline__ v8f wmma_f32_k4(v2f a, v2f b, v8f c) {
  // D = A(16x4) * B(4x16) + C, all f32 (exact f32 semantics, matches reference)
  return __builtin_amdgcn_wmma_f32_16x16x4_f32(false, a, false, b, (short)0, c,
                                               false, false);
}

// ---------------- embedding: h = x @ W_emb + b_emb  (x is [N,4]) -------------
__global__ __launch_bounds__(256) void embed_kernel(
    const float* __restrict__ x, const float* __restrict__ W,
    const float* __restrict__ b, float* __restrict__ h, int N) {
  int idx = blockIdx.x * 256 + threadIdx.x;
  if (idx >= N * D) return;
  int n = idx >> 6, d = idx & 63;
  const float4 xv = *(const float4*)(x + n * 4);
  h[idx] = b[d] + xv.x * W[d] + xv.y * W[D + d] + xv.z * W[2 * D + d] +
           xv.w * W[3 * D + d];
}

// ------------- fused dual GEMM: Oa = X@Wa + ba ; Ob = X@Wb  ------------------
// One wave computes one 16x16 output tile for BOTH weight matrices,
// sharing the A-operand loads of X (reads h once instead of twice).
__global__ __launch_bounds__(256) void gemm_dual(
    const float* __restrict__ X, const float* __restrict__ Wa,
    const float* __restrict__ ba, const float* __restrict__ Wb,
    float* __restrict__ Oa, float* __restrict__ Ob, int nTiles) {
  int wave = (blockIdx.x * 256 + threadIdx.x) >> 5;
  int rt = wave >> 2;                 // row tile (16 rows)
  if (rt >= nTiles) return;           // uniform per wave -> EXEC stays all-ones
  int ct = wave & 3;                  // col tile (16 of 64 cols)
  int lane = threadIdx.x & 31;
  int half = lane >> 4;
  int lr = lane & 15;
  int row = rt * 16 + lr;
  int col = ct * 16 + lr;

  v8f acc1, acc2;
  float bv = ba[col];
#pragma unroll
  for (int v = 0; v < 8; ++v) { acc1[v] = bv; acc2[v] = 0.0f; }

  const float* xr = X + row * D + 2 * half;   // A layout: lanes16-31 hold K+2,K+3
#pragma unroll
  for (int s = 0; s < 16; ++s) {
    int ka = 4 * s + 2 * half;
    v2f aop = *(const v2f*)(xr + 4 * s);
    v2f b1op, b2op;
    b1op.x = Wa[ka * D + col];
    b1op.y = Wa[(ka + 1) * D + col];
    b2op.x = Wb[ka * D + col];
    b2op.y = Wb[(ka + 1) * D + col];
    acc1 = wmma_f32_k4(aop, b1op, acc1);
    acc2 = wmma_f32_k4(aop, b2op, acc2);
  }
#pragma unroll
  for (int v = 0; v < 8; ++v) {
    int orow = rt * 16 + v + 8 * half;        // C/D layout: vgpr v -> row v / v+8
    Oa[orow * D + col] = acc1[v];
    Ob[orow * D + col] = acc2[v];
  }
}

// ------------- accumulate GEMM: O = O + X@Wm + bias  -------------------------
// Existing agg tile is loaded straight into the WMMA C operand.
__global__ __launch_bounds__(256) void gemm_acc(
    const float* __restrict__ X, const float* __restrict__ Wm,
    const float* __restrict__ bias, float* __restrict__ O, int nTiles) {
  int wave = (blockIdx.x * 256 + threadIdx.x) >> 5;
  int rt = wave >> 2;
  if (rt >= nTiles) return;
  int ct = wave & 3;
  int lane = threadIdx.x & 31;
  int half = lane >> 4;
  int lr = lane & 15;
  int row = rt * 16 + lr;
  int col = ct * 16 + lr;

  v8f acc;
  float bv = bias[col];
#pragma unroll
  for (int v = 0; v < 8; ++v)
    acc[v] = O[(rt * 16 + v + 8 * half) * D + col] + bv;

  const float* xr = X + row * D + 2 * half;
#pragma unroll
  for (int s = 0; s < 16; ++s) {
    int ka = 4 * s + 2 * half;
    v2f aop = *(const v2f*)(xr + 4 * s);
    v2f bop;
    bop.x = Wm[ka * D + col];
    bop.y = Wm[(ka + 1) * D + col];
    acc = wmma_f32_k4(aop, bop, acc);
  }
#pragma unroll
  for (int v = 0; v < 8; ++v)
    O[(rt * 16 + v + 8 * half) * D + col] = acc[v];
}

// ------------- edge scatter: agg[dst] += attr * (a[src] - b[dst]) ------------
__global__ __launch_bounds__(256) void edge_kernel(
    const float* __restrict__ A, const float* __restrict__ Bm,
    const float* __restrict__ attr, const int* __restrict__ eid,
    float* __restrict__ agg, int E) {
  int tid = blockIdx.x * 256 + threadIdx.x;
  int e = tid >> 4;
  if (e >= E) return;
  int d0 = (tid & 15) * 4;
  int s = eid[e];
  int d = eid[E + e];
  float w = attr[e];
  const float4 av = *(const float4*)(A + s * D + d0);
  const float4 bv = *(const float4*)(Bm + d * D + d0);
  float* p = agg + d * D + d0;
  atomicAdd(p + 0, w * (av.x - bv.x));
  atomicAdd(p + 1, w * (av.y - bv.y));
  atomicAdd(p + 2, w * (av.z - bv.z));
  atomicAdd(p + 3, w * (av.w - bv.w));
}

// ------------- BatchNorm statistics (sum, sumsq per column) ------------------
__global__ __launch_bounds__(256) void bn_stats(const float* __restrict__ X,
                                                float* __restrict__ stats,
                                                int N) {
  int col = threadIdx.x & 63;
  int slot = blockIdx.x * 4 + (threadIdx.x >> 6);
  int stride = gridDim.x * 4;
  float s = 0.f, s2 = 0.f;
  for (int r = slot; r < N; r += stride) {
    float v = X[r * D + col];
    s += v;
    s2 += v * v;
  }
  atomicAdd(&stats[col], s);
  atomicAdd(&stats[D + col], s2);
}

// ------------- BN apply + ReLU ----------------------------------------------
__global__ __launch_bounds__(256) void bn_apply(
    const float* __restrict__ X, const float* __restrict__ stats,
    const float* __restrict__ gamma, const float* __restrict__ beta,
    float* __restrict__ Y, int N) {
  int idx = blockIdx.x * 256 + threadIdx.x;
  if (idx >= N * D) return;
  int d = idx & 63;
  float invN = 1.0f / (float)N;
  float mean = stats[d] * invN;
  float var = stats[D + d] * invN - mean * mean;
  float sc = gamma[d] * rsqrtf(var + 1e-5f);
  float v = (X[idx] - mean) * sc + beta[d];
  Y[idx] = fmaxf(v, 0.0f);
}

// ------------- global mean pool ---------------------------------------------
__global__ __launch_bounds__(256) void pool_sum(const float* __restrict__ h,
                                                const int* __restrict__ batch,
                                                float* __restrict__ pool,
                                                int N) {
  int tid = blockIdx.x * 256 + threadIdx.x;
  int n = tid >> 4;
  if (n >= N) return;
  int d0 = (tid & 15) * 4;
  int g = batch[n];
  const float4 v = *(const float4*)(h + n * D + d0);
  float* p = pool + g * D + d0;
  atomicAdd(p + 0, v.x);
  atomicAdd(p + 1, v.y);
  atomicAdd(p + 2, v.z);
  atomicAdd(p + 3, v.w);
}

__global__ __launch_bounds__(256) void pool_count(const int* __restrict__ batch,
                                                  float* __restrict__ counts,
                                                  int N) {
  int n = blockIdx.x * 256 + threadIdx.x;
  if (n >= N) return;
  atomicAdd(&counts[batch[n]], 1.0f);
}

// ------------- MLP head: pred = relu(g@Wl1+bl1)@Wl2 + bl2 --------------------
__global__ __launch_bounds__(256) void head_kernel(
    const float* __restrict__ pool, const float* __restrict__ counts,
    const float* __restrict__ Wl1, const float* __restrict__ bl1,
    const float* __restrict__ Wl2, const float* __restrict__ bl2,
    float* __restrict__ out) {
  int g = threadIdx.x;  // 256 graphs, one thread each
  float inv = 1.0f / fmaxf(counts[g], 1.0f);
  float gv[D];
#pragma unroll
  for (int d = 0; d < D; ++d) gv[d] = pool[g * D + d] * inv;
  float o0 = bl2[0], o1 = bl2[1], o2 = bl2[2];
  for (int j = 0; j < D; ++j) {
    float hs = bl1[j];
#pragma unroll
    for (int d = 0; d < D; ++d) hs += gv[d] * Wl1[d * D + j];
    hs = fmaxf(hs, 0.0f);
    o0 += hs * Wl2[j * 3 + 0];
    o1 += hs * Wl2[j * 3 + 1];
    o2 += hs * Wl2[j * 3 + 2];
  }
  out[g * 3 + 0] = o0;
  out[g * 3 + 1] = o1;
  out[g * 3 + 2] = o2;
}

extern "C" void kernel_launch(void* const* d_in, const int* in_sizes, int n_in,
                              void* d_out, int out_size, void* d_ws,
                              size_t ws_size, hipStream_t stream) {
  const float* x       = (const float*)d_in[0];
  const int*   EdgeID  = (const int*)d_in[1];
  const float* EdgeAttr= (const float*)d_in[2];
  const int*   batch   = (const int*)d_in[3];
  const float* W_emb   = (const float*)d_in[4];
  const float* b_emb   = (const float*)d_in[5];
  const float* W1      = (const float*)d_in[6];
  const float* b1      = (const float*)d_in[7];
  const float* W2      = (const float*)d_in[8];
  const float* W3      = (const float*)d_in[9];
  const float* b3      = (const float*)d_in[10];
  const float* gamma   = (const float*)d_in[11];
  const float* beta    = (const float*)d_in[12];
  const float* W_l1    = (const float*)d_in[13];
  const float* b_l1    = (const float*)d_in[14];
  const float* W_l2    = (const float*)d_in[15];
  const float* b_l2    = (const float*)d_in[16];

  const int N = in_sizes[0] / 4;   // x is [N,4]
  const int E = in_sizes[2];       // EdgeAttr is [E]
  const int L = in_sizes[7] / D;   // b1 is [L,64]

  // workspace layout (floats): h | hn(agg) | a | b | stats(128) | pool | counts
  float* ws = (float*)d_ws;
  size_t N64 = (size_t)N * D;
  float* h      = ws;
  float* hn     = ws + N64;
  float* abuf   = ws + 2 * N64;
  float* bbuf   = ws + 3 * N64;
  float* stats  = ws + 4 * N64;
  float* pool   = stats + 2 * D;
  float* counts = pool + NGRAPH * D;

  const int nTiles = (N + 15) / 16;              // N=100000 -> exact 6250
  const int gemmBlocks = (nTiles * 4 + 7) / 8;   // 8 waves / block
  dim3 blk(256);

  embed_kernel<<<(N * D + 255) / 256, blk, 0, stream>>>(x, W_emb, b_emb, h, N);

  for (int i = 0; i < L; ++i) {
    gemm_dual<<<gemmBlocks, blk, 0, stream>>>(h, W1 + i * D * D, b1 + i * D,
                                              W2 + i * D * D, abuf, bbuf,
                                              nTiles);
    hipMemsetAsync(hn, 0, N64 * sizeof(float), stream);
    edge_kernel<<<(E * 16 + 255) / 256, blk, 0, stream>>>(abuf, bbuf, EdgeAttr,
                                                          EdgeID, hn, E);
    gemm_acc<<<gemmBlocks, blk, 0, stream>>>(h, W3 + i * D * D, b3 + i * D, hn,
                                             nTiles);
    hipMemsetAsync(stats, 0, 2 * D * sizeof(float), stream);
    bn_stats<<<1024, blk, 0, stream>>>(hn, stats, N);
    bn_apply<<<(N * D + 255) / 256, blk, 0, stream>>>(hn, stats, gamma + i * D,
                                                      beta + i * D, h, N);
  }

  hipMemsetAsync(pool, 0, (NGRAPH * D + NGRAPH) * sizeof(float), stream);
  pool_sum<<<(N * 16 + 255) / 256, blk, 0, stream>>>(h, batch, pool, N);
  pool_count<<<(N + 255) / 256, blk, 0, stream>>>(batch, counts, N);
  head_kernel<<<1, 256, 0, stream>>>(pool, counts, W_l1, b_l1, W_l2, b_l2,
                                     (float*)d_out);
}